// FP8Linear_26989574488647
// MI455X (gfx1250) — compile-verified
//
#include <hip/hip_runtime.h>

typedef __attribute__((ext_vector_type(2))) float v2f;
typedef __attribute__((ext_vector_type(8))) float v8f;

static constexpr int MDIM  = 32;
static constexpr int IDIM  = 7168;
static constexpr int ODIM  = 18432;
static constexpr int SBLK  = 128;
static constexpr int SCOLS = IDIM / SBLK;        // 56 scale columns
static constexpr int SPLITK  = 7;                // K-splits for occupancy
static constexpr int KSPLIT  = IDIM / SPLITK;    // 1024 K per split
static constexpr int KCHUNK  = 256;              // K per LDS stage (2 scale blocks)
static constexpr int KSTRIDE = 260;              // padded LDS row stride (260 % 64 == 4 -> conflict-free)
static constexpr int NSTAGE  = KSPLIT / KCHUNK;  // 4
static constexpr int WG_OROWS = 128;             // 8 waves x 16 O-rows
static constexpr int NOBLK    = ODIM / WG_OROWS; // 144

// Each wave: 32(M) x 16(O) output tile via two v_wmma_f32_16x16x4_f32 per K-step of 4.
template <bool USE_WS>
__global__ __launch_bounds__(256)
void fp8lin_gemm(const float* __restrict__ x,     // [32, 7168]
                 const float* __restrict__ w,     // [18432, 7168]
                 const float* __restrict__ s,     // [144, 56]
                 float* __restrict__ part,        // [SPLITK, 32, 18432] (USE_WS)
                 float* __restrict__ out)         // [32, 18432] (!USE_WS, pre-zeroed)
{
    __shared__ float xs[MDIM * KSTRIDE];          // 33.3 KB

    const int oblk   = blockIdx.x;                // 0..143
    const int ksplit = blockIdx.y;                // 0..6
    const int tid    = threadIdx.x;
    const int wave   = tid >> 5;
    const int lane   = tid & 31;
    const int n      = lane & 15;                 // N column within tile / M row for A
    const int kk     = (lane >> 4) << 1;          // K sub-offset per A/B VGPR layout: 0 or 2

    const int o_base = oblk * WG_OROWS + wave * 16;
    const float* __restrict__ wrow = w + (size_t)(o_base + n) * IDIM;

    v8f c0 = {};  // M rows 0..15
    v8f c1 = {};  // M rows 16..31

    const int mA0 = n;
    const int mA1 = 16 + n;

    for (int stage = 0; stage < NSTAGE; ++stage) {
        const int kg = ksplit * KSPLIT + stage * KCHUNK;   // global K base of this stage
        __syncthreads();                                   // protect previous stage's readers
        // Coalesced cooperative stage: x[0:32, kg:kg+256] -> padded LDS rows.
        #pragma unroll
        for (int idx = tid; idx < MDIM * KCHUNK; idx += 256) {
            const int m  = idx >> 8;       // idx / 256
            const int kl = idx & 255;      // idx % 256
            xs[m * KSTRIDE + kl] = x[(size_t)m * IDIM + kg + kl];
        }
        __syncthreads();

        #pragma unroll
        for (int half = 0; half < 2; ++half) {             // one scale block per 128 K
            const float sv = s[(size_t)oblk * SCOLS + ((kg >> 7) + half)];
            const int kb = half * 128;
            #pragma unroll 8
            for (int k4 = 0; k4 < 128; k4 += 4) {
                const int koff = kb + k4 + kk;
                // A (16x4 f32): lane m = n (+16 for c1), two contiguous K values -> ds_load_b64
                const v2f a0 = *(const v2f*)&xs[mA0 * KSTRIDE + koff];
                const v2f a1 = *(const v2f*)&xs[mA1 * KSTRIDE + koff];
                // B (4x16 f32): lane holds W[o_base+n][k..k+1], dequant folded in
                v2f b = *(const v2f*)(wrow + kg + koff);
                b.x *= sv;
                b.y *= sv;
                c0 = __builtin_amdgcn_wmma_f32_16x16x4_f32(false, a0, false, b,
                                                           (short)0, c0, false, false);
                c1 = __builtin_amdgcn_wmma_f32_16x16x4_f32(false, a1, false, b,
                                                           (short)0, c1, false, false);
            }
        }
    }

    // C/D layout: VGPR j -> M = j + (lane<16 ? 0 : 8), N = lane & 15
    const int mlow = (lane < 16) ? 0 : 8;
    const int ocol = o_base + n;
    #pragma unroll
    for (int j = 0; j < 8; ++j) {
        const int m0 = mlow + j;
        const int m1 = 16 + mlow + j;
        if (USE_WS) {
            part[((size_t)ksplit * MDIM + m0) * ODIM + ocol] = c0[j];
            part[((size_t)ksplit * MDIM + m1) * ODIM + ocol] = c1[j];
        } else {
            atomicAdd(&out[(size_t)m0 * ODIM + ocol], c0[j]);
            atomicAdd(&out[(size_t)m1 * ODIM + ocol], c1[j]);
        }
    }
}

__global__ void fp8lin_reduce(const float* __restrict__ part, float* __restrict__ out)
{
    const size_t nelem = (size_t)MDIM * ODIM;
    const size_t i = (size_t)blockIdx.x * blockDim.x + threadIdx.x;
    if (i < nelem) {
        float acc = 0.f;
        #pragma unroll
        for (int sp = 0; sp < SPLITK; ++sp)
            acc += part[(size_t)sp * nelem + i];
        out[i] = acc;
    }
}

__global__ void fp8lin_zero(float* __restrict__ out)
{
    const size_t nelem = (size_t)MDIM * ODIM;
    const size_t i = (size_t)blockIdx.x * blockDim.x + threadIdx.x;
    if (i < nelem) out[i] = 0.f;
}

extern "C" void kernel_launch(void* const* d_in, const int* in_sizes, int n_in,
                              void* d_out, int out_size, void* d_ws, size_t ws_size,
                              hipStream_t stream)
{
    const float* x = (const float*)d_in[0];
    const float* w = (const float*)d_in[1];
    const float* s = (const float*)d_in[2];
    float* out = (float*)d_out;

    const size_t nelem = (size_t)MDIM * ODIM;
    const size_t need  = (size_t)SPLITK * nelem * sizeof(float);
    const dim3 grid(NOBLK, SPLITK);
    const int rblocks = (int)((nelem + 255) / 256);

    if (ws_size >= need) {
        // Deterministic path: split-K partials in workspace + sequential per-element sum.
        float* part = (float*)d_ws;
        fp8lin_gemm<true><<<grid, 256, 0, stream>>>(x, w, s, part, out);
        fp8lin_reduce<<<rblocks, 256, 0, stream>>>(part, out);
    } else {
        // Fallback: zero output, accumulate split-K with fp32 atomics.
        fp8lin_zero<<<rblocks, 256, 0, stream>>>(out);
        fp8lin_gemm<false><<<grid, 256, 0, stream>>>(x, w, s, nullptr, out);
    }
}